// SpatialEncoding_46943992545790
// MI455X (gfx1250) — compile-verified
//
#include <hip/hip_runtime.h>

// Problem constants (match reference).
#define TT 4
#define NN 50000
#define EE 800000
#define DD 256
// GEMM: wave computes a 16x64 output tile (4 WMMA accumulators).
// Tile groups = (N/16) * (D/64) = 3125 * 4 = 12500 waves ; 4 waves/block -> 3125 blocks.

typedef __attribute__((ext_vector_type(2))) float v2f;
typedef __attribute__((ext_vector_type(8))) float v8f;

// ---------------------------------------------------------------- degree prep
__global__ __launch_bounds__(256) void k_deg_init(float* __restrict__ deg) {
  int i = blockIdx.x * 256 + threadIdx.x;
  if (i < NN) deg[i] = 1.0f;  // self-loop weight
}

__global__ __launch_bounds__(256) void k_deg_scatter(const int* __restrict__ col,
                                                     const float* __restrict__ w,
                                                     float* __restrict__ deg) {
  int e = blockIdx.x * 256 + threadIdx.x;  // E divisible by 256
  __hip_atomic_fetch_add(&deg[col[e]], w[e], __ATOMIC_RELAXED,
                         __HIP_MEMORY_SCOPE_AGENT);
}

__global__ __launch_bounds__(256) void k_dinv(float* __restrict__ deg) {
  int i = blockIdx.x * 256 + threadIdx.x;
  if (i < NN) {
    float d = deg[i];
    deg[i] = (d > 0.0f) ? rsqrtf(d) : 0.0f;  // deg buffer becomes dinv
  }
}

// ---------------------------------------------------------------- embedding
__global__ __launch_bounds__(256) void k_embed(const int* __restrict__ ids,
                                               const float4* __restrict__ emb4,
                                               float4* __restrict__ x4) {
  int gid = blockIdx.x * 256 + threadIdx.x;  // over N*64 float4's (exact)
  int i = gid >> 6;
  int j = gid & 63;
  x4[gid] = emb4[(size_t)ids[i] * 64 + j];
}

// ---------------------------------------------------------------- fp32 WMMA GEMM
// H[N,256] = X[N,256] @ W[256,256]. Each wave: one 16x64 tile via 4 independent
// V_WMMA_F32_16X16X4_F32 accumulator chains; A fragment loaded once per K-step
// and reused across the 4 N-tiles (4x lower A traffic, 4-way WMMA ILP).
__global__ __launch_bounds__(128) void k_gemm_wmma(const float* __restrict__ X,
                                                   const float* __restrict__ Wm,
                                                   float* __restrict__ H) {
  int wave = threadIdx.x >> 5;
  int lane = threadIdx.x & 31;
  int tg = blockIdx.x * 4 + wave;    // 12500 tile groups, exact
  int mt = tg >> 2;                  // 0..3124 : 16-row block
  int ng = tg & 3;                   // 0..3    : 64-col block
  int half = lane >> 4;              // 0|1 : K sub-slot (A/B) and M+8 (C/D)
  int l16 = lane & 15;               // A: M row ; B/C/D: N column

  const float* arow = X + (size_t)(mt * 16 + l16) * DD;   // A: M = l16
  const float* bbase = Wm + ng * 64 + l16;                // B: N base = ng*64+l16

  v8f acc[4] = {};
#pragma unroll 4
  for (int k0 = 0; k0 < DD; k0 += 4) {
    int ka = k0 + (half << 1);  // lanes 0-15: K=k0,k0+1 ; lanes 16-31: K=k0+2,k0+3
    v2f a;
    a.x = arow[ka];
    a.y = arow[ka + 1];
#pragma unroll
    for (int j = 0; j < 4; ++j) {
      const float* bcol = bbase + j * 16;
      v2f b;
      b.x = bcol[(size_t)ka * DD];
      b.y = bcol[(size_t)(ka + 1) * DD];
      acc[j] = __builtin_amdgcn_wmma_f32_16x16x4_f32(false, a, false, b,
                                                     (short)0, acc[j], false,
                                                     false);
    }
  }

  // C/D layout: VGPR r, lanes 0-15 -> M=r ; lanes 16-31 -> M=r+8 ; N=l16.
  float* hbase = H + (size_t)(mt * 16) * DD + ng * 64 + l16;
#pragma unroll
  for (int j = 0; j < 4; ++j) {
    float* hp = hbase + j * 16;
#pragma unroll
    for (int r = 0; r < 8; ++r) hp[(size_t)(r + half * 8) * DD] = acc[j][r];
  }
}

// ------------------------------------------------- out = bias + h * dinv^2 (self loop)
__global__ __launch_bounds__(256) void k_out_init(const float4* __restrict__ h4,
                                                  const float* __restrict__ dinv,
                                                  const float4* __restrict__ b4,
                                                  float4* __restrict__ out4) {
  int gid = blockIdx.x * 256 + threadIdx.x;  // over N*64 float4's (exact)
  int i = gid >> 6;
  int j = gid & 63;
  float di = dinv[i];
  float s = di * di;  // self-loop norm = 1/deg
  float4 hv = h4[gid];
  float4 bv = b4[j];
  float4 o;
  o.x = bv.x + s * hv.x;
  o.y = bv.y + s * hv.y;
  o.z = bv.z + s * hv.z;
  o.w = bv.w + s * hv.w;
  out4[gid] = o;
}

// ------------------------------------------------- edge scatter-add (atomics)
// out[c,:] += dinv[r]*w*dinv[c] * h[r,:] ; 64 threads/edge, float4 gather,
// 4 native fp32 atomics per thread (no return -> global_atomic_add_f32).
__global__ __launch_bounds__(256) void k_edge_scatter(const int* __restrict__ row,
                                                      const int* __restrict__ col,
                                                      const float* __restrict__ w,
                                                      const float* __restrict__ dinv,
                                                      const float* __restrict__ h,
                                                      float* __restrict__ out) {
  int e = blockIdx.x * 4 + (threadIdx.x >> 6);  // E divisible by 4
  int d4 = (threadIdx.x & 63) << 2;
  int r = row[e];
  int c = col[e];
  float nrm = dinv[r] * w[e] * dinv[c];
  float4 hv = *(const float4*)(h + (size_t)r * DD + d4);
  float* o = out + (size_t)c * DD + d4;
  __hip_atomic_fetch_add(o + 0, nrm * hv.x, __ATOMIC_RELAXED, __HIP_MEMORY_SCOPE_AGENT);
  __hip_atomic_fetch_add(o + 1, nrm * hv.y, __ATOMIC_RELAXED, __HIP_MEMORY_SCOPE_AGENT);
  __hip_atomic_fetch_add(o + 2, nrm * hv.z, __ATOMIC_RELAXED, __HIP_MEMORY_SCOPE_AGENT);
  __hip_atomic_fetch_add(o + 3, nrm * hv.w, __ATOMIC_RELAXED, __HIP_MEMORY_SCOPE_AGENT);
}

// ---------------------------------------------------------------- driver
extern "C" void kernel_launch(void* const* d_in, const int* in_sizes, int n_in,
                              void* d_out, int out_size, void* d_ws, size_t ws_size,
                              hipStream_t stream) {
  const int* nf = (const int*)d_in[0];      // [T,N]
  const int* ei = (const int*)d_in[1];      // [T,2,E]
  const float* ew = (const float*)d_in[2];  // [T,E]
  const float* emb = (const float*)d_in[3]; // [VOCAB,D]
  const float* Wl[3] = {(const float*)d_in[4], (const float*)d_in[6],
                        (const float*)d_in[8]};
  const float* bl[3] = {(const float*)d_in[5], (const float*)d_in[7],
                        (const float*)d_in[9]};
  float* outp = (float*)d_out;

  const size_t ND = (size_t)NN * DD;
  float* xA = (float*)d_ws;       // [N,D]
  float* xB = xA + ND;            // [N,D]
  float* h = xB + ND;             // [N,D]
  float* dinv = h + ND;           // [N]   (deg, then dinv)

  const int blkN = (NN + 255) / 256;        // N-sized guarded kernels
  const int blkND4 = (NN * (DD / 4)) / 256; // 12500 exact
  const int blkDegE = EE / 256;             // 3125 exact
  const int blkGemm = (NN / 16) * (DD / 64) / 4;  // 3125 exact (128-thread blocks)
  const int blkEdge = EE / 4;               // 200000 exact

  for (int t = 0; t < TT; ++t) {
    const int* ids = nf + (size_t)t * NN;
    const int* row = ei + (size_t)t * 2 * EE;
    const int* col = row + EE;
    const float* w = ew + (size_t)t * EE;

    // Normalization terms (shared across the 3 layers of this timestep).
    k_deg_init<<<blkN, 256, 0, stream>>>(dinv);
    k_deg_scatter<<<blkDegE, 256, 0, stream>>>(col, w, dinv);
    k_dinv<<<blkN, 256, 0, stream>>>(dinv);

    // x = emb[ids]
    k_embed<<<blkND4, 256, 0, stream>>>(ids, (const float4*)emb, (float4*)xA);

    for (int l = 0; l < 3; ++l) {
      const float* xin = (l == 1) ? xB : xA;
      float* xout = (l == 0) ? xB : (l == 1) ? xA : (outp + (size_t)t * ND);

      k_gemm_wmma<<<blkGemm, 128, 0, stream>>>(xin, Wl[l], h);
      k_out_init<<<blkND4, 256, 0, stream>>>((const float4*)h, dinv,
                                             (const float4*)bl[l],
                                             (float4*)xout);
      k_edge_scatter<<<blkEdge, 256, 0, stream>>>(row, col, w, dinv, h, xout);
    }
  }
}